// Q6Module_55851754717358
// MI455X (gfx1250) — compile-verified
//
#include <hip/hip_runtime.h>
#include <hip/hip_bf16.h>
#include <float.h>

// ---------------------------------------------------------------------------
// Q6 order parameter, MI455X (gfx1250, wave32).
//
// Distance matrix as WMMA:  d^2(j,i) = |pj|^2 + |pi|^2 - 2 pj.pi
//   A (16x4, M=j):  (x_j, y_j, z_j, |pj|^2)
//   B (4x16, N=i):  (-2x_i, -2y_i, -2z_i, 1)
//   C (16x16)    :  |pi|^2 broadcast per column
//   D = A*B + C  :  one V_WMMA_F32_16X16X4_F32 per 16x16 tile, full f32.
//
// Round-3 changes vs round-2:
//  * self-distance masking moved behind a *uniform* branch: all lanes of a
//    wave have their i-atom in the same 16-aligned j-tile (jb == iBase), so
//    the 8x v_cmp_ne + 8x v_cndmask per tile becomes one s_cmp per tile plus
//    a once-per-512-tiles masking block.
// ---------------------------------------------------------------------------

#define N_ATOMS         8192
#define NUM_NBS         16
#define WAVES_PER_BLOCK 4
#define BLOCK_THREADS   (WAVES_PER_BLOCK * 32)
#define ATOMS_PER_BLOCK (WAVES_PER_BLOCK * 16)
#define NUM_BLOCKS      (N_ATOMS / ATOMS_PER_BLOCK)

typedef float v2f __attribute__((ext_vector_type(2)));
typedef float v8f __attribute__((ext_vector_type(8)));

__global__ void q6_zero_kernel(float* out) { out[0] = 0.0f; }

__launch_bounds__(BLOCK_THREADS)
__global__ void q6_kernel(const float* __restrict__ pos, float* __restrict__ out) {
    // Entire position set staged in LDS: 3 * 8192 * 4B = 96 KB (< 320 KB/WGP).
    __shared__ float lX[N_ATOMS];
    __shared__ float lY[N_ATOMS];
    __shared__ float lZ[N_ATOMS];

    const int tid = threadIdx.x;
    for (int t = tid; t < N_ATOMS; t += BLOCK_THREADS) {
        lX[t] = pos[3 * t + 0];
        lY[t] = pos[3 * t + 1];
        lZ[t] = pos[3 * t + 2];
    }
    __syncthreads();

    const int  lane = tid & 31;
    const int  wave = tid >> 5;
    const int  im   = lane & 15;          // this lane's i-atom within the wave tile
    const bool hiHalf = (lane >= 16);
    const int  iBase   = blockIdx.x * ATOMS_PER_BLOCK + wave * 16;  // uniform
    const int  iGlobal = iBase + im;

    const float xi = lX[iGlobal], yi = lY[iGlobal], zi = lZ[iGlobal];
    const float sqi = xi * xi + yi * yi + zi * zi;

    // B matrix (i-atoms): K=0,1 on lanes 0-15, K=2,3 on lanes 16-31.
    v2f b;
    b.x = hiHalf ? (-2.0f * zi) : (-2.0f * xi);   // K=0 -> -2x, K=2 -> -2z
    b.y = hiHalf ? 1.0f         : (-2.0f * yi);   // K=1 -> -2y, K=3 -> 1
    const v8f cmat = {sqi, sqi, sqi, sqi, sqi, sqi, sqi, sqi};

    // Per-lane top-16 smallest d^2 (registers only: all indices static after
    // unroll; worst value + worst slot tracked to make inserts O(16) cndmasks).
    float bd[NUM_NBS];
    int   bi[NUM_NBS];
#pragma unroll
    for (int s = 0; s < NUM_NBS; ++s) { bd[s] = FLT_MAX; bi[s] = 0; }
    float worst = FLT_MAX;
    int   wpos  = 0;

    const int rowOff = hiHalf ? 8 : 0;

    // ---- software-pipelined tile loop -------------------------------------
    // Registers hold the j-tile currently being consumed; loads for the next
    // tile are issued before the WMMA so the DS wait overlaps compute.
    float xj = lX[im], yj = lY[im], zj = lZ[im];

    for (int jb = 0; jb < N_ATOMS; jb += 16) {
        // issue next-tile loads now (wrap-around index: no branch in loop)
        const int jn = ((jb + 16) & (N_ATOMS - 1)) + im;
        const float nx = lX[jn];
        const float ny = lY[jn];
        const float nz = lZ[jn];

        const float sqj = xj * xj + yj * yj + zj * zj;
        v2f a;
        a.x = hiHalf ? zj  : xj;   // K=0 -> x_j, K=2 -> z_j
        a.y = hiHalf ? sqj : yj;   // K=1 -> y_j, K=3 -> |p_j|^2

        // Uniform control flow here: EXEC all ones as WMMA requires.
        v8f d2 = __builtin_amdgcn_wmma_f32_16x16x4_f32(
            /*neg_a=*/false, a, /*neg_b=*/false, b,
            /*c_mod=*/(short)0, cmat, /*reuse_a=*/false, /*reuse_b=*/false);

        float cand[8];
#pragma unroll
        for (int r = 0; r < 8; ++r) cand[r] = d2[r];

        // Self-distance can only appear when the j-tile IS the wave's i-tile
        // (iGlobal = iBase + im, all lanes share the 16-aligned iBase), so
        // this is a uniform, once-per-512-tiles branch.
        if (jb == iBase) {
#pragma unroll
            for (int r = 0; r < 8; ++r)
                if (rowOff + r == im) cand[r] = FLT_MAX;   // jg == iGlobal
        }

        const float m01 = fminf(cand[0], cand[1]);
        const float m23 = fminf(cand[2], cand[3]);
        const float m45 = fminf(cand[4], cand[5]);
        const float m67 = fminf(cand[6], cand[7]);
        const float mall = fminf(fminf(m01, m23), fminf(m45, m67));

        if (mall < worst) {                        // rare after warm-up
#pragma unroll
            for (int r = 0; r < 8; ++r) {
                const float dv = cand[r];
                if (dv < worst) {
                    const int jg = jb + rowOff + r;
#pragma unroll
                    for (int s = 0; s < NUM_NBS; ++s)
                        if (s == wpos) { bd[s] = dv; bi[s] = jg; }
                    worst = bd[0]; wpos = 0;
#pragma unroll
                    for (int s = 1; s < NUM_NBS; ++s)
                        if (bd[s] > worst) { worst = bd[s]; wpos = s; }
                }
            }
        }

        xj = nx; yj = ny; zj = nz;                 // advance pipeline
    }

    // Lanes L and L+16 hold partial top-16 lists for the same i-atom over
    // disjoint halves of j.  True top-16 of the union is contained in the two
    // partial lists; merge into lanes 0-15 via cross-half shuffles.
#pragma unroll
    for (int s = 0; s < NUM_NBS; ++s) {
        const float od = __shfl_xor(bd[s], 16, 32);
        const int   oi = __shfl_xor(bi[s], 16, 32);
        if (!hiHalf && od < worst) {
#pragma unroll
            for (int t = 0; t < NUM_NBS; ++t)
                if (t == wpos) { bd[t] = od; bi[t] = oi; }
            worst = bd[0]; wpos = 0;
#pragma unroll
            for (int t = 1; t < NUM_NBS; ++t)
                if (bd[t] > worst) { worst = bd[t]; wpos = t; }
        }
    }

    if (hiHalf) return;  // phase 2 on lanes 0-15 only (one lane per i-atom)

    // ---- Phase 2: l=6 real SH sums over the 16 neighbors -------------------
    const float SQ2 = 1.41421356237f;
    // N_m = sqrt(13/(4pi) * (6-m)!/(6+m)!)
    const float N0 = 1.0171078f,  N1 = 0.15694305f, N2 = 0.02481488f,
                N3 = 0.00413583f, N4 = 7.550926e-4f, N5 = 1.609864e-4f,
                N6 = 4.647270e-5f;

    float acc[13];
#pragma unroll
    for (int k = 0; k < 13; ++k) acc[k] = 0.0f;
    float ssum = 0.0f;

#pragma unroll
    for (int s = 0; s < NUM_NBS; ++s) {
        const int   j  = bi[s];
        const float dx = lX[j] - xi, dy = lY[j] - yi, dz = lZ[j] - zi;
        const float dist = sqrtf(dx * dx + dy * dy + dz * dz);
        const float inv  = 1.0f / dist;
        const float ux = dx * inv, uy = dy * inv, uz = dz * inv;

        const float dd  = dist - 0.05f;                 // D0
        const float sig = __expf(-dd * dd * 800.0f);    // 1/(2*R0^2) = 800

        const float z2 = uz * uz, z4 = z2 * z2;
        const float i16 = 0.0625f;
        const float Q0 = (231.0f * z4 * z2 - 315.0f * z4 + 105.0f * z2 - 5.0f) * i16;
        const float Q1 = (1386.0f * z4 - 1260.0f * z2 + 210.0f) * uz * i16;
        const float Q2 = (6930.0f * z4 - 3780.0f * z2 + 210.0f) * i16;
        const float Q3 = (27720.0f * z2 - 7560.0f) * uz * i16;
        const float Q4 = (83160.0f * z2 - 7560.0f) * i16;
        const float Q5 = 10395.0f * uz;
        const float Q6 = 10395.0f;

        const float c1 = ux,                 s1 = uy;
        const float c2 = ux * c1 - uy * s1,  s2 = ux * s1 + uy * c1;
        const float c3 = ux * c2 - uy * s2,  s3 = ux * s2 + uy * c2;
        const float c4 = ux * c3 - uy * s3,  s4 = ux * s3 + uy * c3;
        const float c5 = ux * c4 - uy * s4,  s5 = ux * s4 + uy * c4;
        const float c6 = ux * c5 - uy * s5,  s6 = ux * s5 + uy * c5;

        acc[0]  += sig * (SQ2 * N6 * Q6 * s6);
        acc[1]  += sig * (SQ2 * N5 * Q5 * s5);
        acc[2]  += sig * (SQ2 * N4 * Q4 * s4);
        acc[3]  += sig * (SQ2 * N3 * Q3 * s3);
        acc[4]  += sig * (SQ2 * N2 * Q2 * s2);
        acc[5]  += sig * (SQ2 * N1 * Q1 * s1);
        acc[6]  += sig * (N0 * Q0);
        acc[7]  += sig * (SQ2 * N1 * Q1 * c1);
        acc[8]  += sig * (SQ2 * N2 * Q2 * c2);
        acc[9]  += sig * (SQ2 * N3 * Q3 * c3);
        acc[10] += sig * (SQ2 * N4 * Q4 * c4);
        acc[11] += sig * (SQ2 * N5 * Q5 * c5);
        acc[12] += sig * (SQ2 * N6 * Q6 * c6);
        ssum += sig;
    }

    float s2sum = 0.0f;
#pragma unroll
    for (int k = 0; k < 13; ++k) s2sum += acc[k] * acc[k];
    const float qi = sqrtf(s2sum) / ssum;   // sqrt(sum q_nm^2)/sum(sig)

    atomicAdd(out, qi * (1.0f / (float)N_ATOMS));
}

extern "C" void kernel_launch(void* const* d_in, const int* in_sizes, int n_in,
                              void* d_out, int out_size, void* d_ws, size_t ws_size,
                              hipStream_t stream) {
    (void)in_sizes; (void)n_in; (void)out_size; (void)d_ws; (void)ws_size;
    const float* pos = (const float*)d_in[0];
    float* out = (float*)d_out;
    q6_zero_kernel<<<1, 1, 0, stream>>>(out);
    q6_kernel<<<NUM_BLOCKS, BLOCK_THREADS, 0, stream>>>(pos, out);
}